// Attention_37847251812733
// MI455X (gfx1250) — compile-verified
//
#include <hip/hip_runtime.h>

typedef float v2f __attribute__((ext_vector_type(2)));
typedef float v8f __attribute__((ext_vector_type(8)));

#if defined(__HIP_DEVICE_COMPILE__)
#if !__has_builtin(__builtin_amdgcn_wmma_f32_16x16x4_f32)
#error "DEVICE: wmma f32 16x16x4 builtin not available on this toolchain"
#endif
#endif

__device__ __forceinline__ v8f wmma_f32(v2f a, v2f b, v8f c) {
    // 8 args: (neg_a, A, neg_b, B, c_mod, C, reuse_a, reuse_b)
    return __builtin_amdgcn_wmma_f32_16x16x4_f32(false, a, false, b, (short)0, c,
                                                 false, false);
}

// Async global -> LDS copy, 16B per lane, tracked with ASYNCcnt.
// LDS byte address = low 32 bits of the generic pointer (flat LDS aperture).
__device__ __forceinline__ void async_copy_b128(void* lds_dst, const void* gsrc) {
    asm volatile("global_load_async_to_lds_b128 %0, %1, off"
                 :
                 : "v"((unsigned)(unsigned long long)lds_dst), "v"(gsrc)
                 : "memory");
}
__device__ __forceinline__ void wait_async_le3() {
    asm volatile("s_wait_asynccnt 0x3" ::: "memory");
}
__device__ __forceinline__ void wait_async_0() {
    asm volatile("s_wait_asynccnt 0x0" ::: "memory");
}

#define KB 16  // k-block depth staged in LDS (4 WMMA k-steps)

// ---------------------------------------------------------------------------
// Y[M x Nout] = X[M x K] @ W[Nout x K]^T  (+ bias, optional)
// 256 threads = 8 waves; WG tile 128(M) x 64(N); wave tile 32x32 = 2x2 WMMA.
// Double-buffered async global->LDS staging (3 async b128 ops per thread
// per tile: 2 for the A tile, 1 for the B tile).
// M % 128 == 0, K % KB == 0, Nout % 64 == 0  (holds for 4096/1024/1024)
// ---------------------------------------------------------------------------
__global__ __launch_bounds__(256) void gemm_xwt_f32(
    const float* __restrict__ X, const float* __restrict__ W,
    const float* __restrict__ bias, float* __restrict__ Y, int M, int K,
    int Nout) {
    __shared__ float As[2][128][KB];  // 16 KB
    __shared__ float Bs[2][64][KB];   // 8 KB

    const int tid  = threadIdx.x;
    const int wave = tid >> 5;
    const int lane = tid & 31;
    const int l16  = lane & 15;
    const int hi   = lane >> 4;
    const int wm   = (wave >> 1) * 32;  // wave M offset: 0..96
    const int wn   = (wave & 1) * 32;   // wave N offset: 0..32
    const int bm   = blockIdx.x * 128;
    const int bn   = blockIdx.y * 64;

    // per-thread staging slots (fixed for all k-blocks)
    const int arow0 = tid >> 2;            // A slot 0: rows 0..63
    const int arow1 = (tid + 256) >> 2;    // A slot 1: rows 64..127
    const int ac4   = (tid & 3) << 2;
    const int brow  = tid >> 2;            // B rows 0..63
    const int bc4   = (tid & 3) << 2;

    auto issue_tile = [&](int buf, int k0) {
        async_copy_b128(&As[buf][arow0][ac4],
                        &X[(size_t)(bm + arow0) * K + k0 + ac4]);
        async_copy_b128(&As[buf][arow1][ac4],
                        &X[(size_t)(bm + arow1) * K + k0 + ac4]);
        async_copy_b128(&Bs[buf][brow][bc4],
                        &W[(size_t)(bn + brow) * K + k0 + bc4]);
    };

    v8f acc[2][2];
    const v8f vzero = {};
    acc[0][0] = vzero; acc[0][1] = vzero; acc[1][0] = vzero; acc[1][1] = vzero;

    issue_tile(0, 0);  // prologue: fill buffer 0

    for (int k0 = 0; k0 < K; k0 += KB) {
        const int cur = (k0 / KB) & 1;
        if (k0 + KB < K) {
            issue_tile(cur ^ 1, k0 + KB);  // overlap next tile with compute
            if (k0 + 2 * KB < K)           // L2 prefetch two tiles ahead
                __builtin_prefetch(&X[(size_t)(bm + arow0) * K + k0 + 2 * KB + ac4],
                                   0, 1);
            wait_async_le3();  // only the 3 just-issued remain outstanding
        } else {
            wait_async_0();
        }
        __syncthreads();  // current buffer fully populated by all waves

        // ---- 4 WMMA k-steps over this k-block ----
        #pragma unroll
        for (int kk = 0; kk < KB; kk += 4) {
            v2f a[2], b[2];
            #pragma unroll
            for (int mt = 0; mt < 2; ++mt)
                a[mt] = *(const v2f*)&As[cur][wm + mt * 16 + l16][kk + 2 * hi];
            #pragma unroll
            for (int nt = 0; nt < 2; ++nt)
                b[nt] = *(const v2f*)&Bs[cur][wn + nt * 16 + l16][kk + 2 * hi];
            #pragma unroll
            for (int mt = 0; mt < 2; ++mt)
                #pragma unroll
                for (int nt = 0; nt < 2; ++nt)
                    acc[mt][nt] = wmma_f32(a[mt], b[nt], acc[mt][nt]);
        }
        __syncthreads();  // all waves done reading before buffer is rewritten
    }

    // ---- epilogue: D-layout scatter (row = v + 8*hi, col = l16) ----
    #pragma unroll
    for (int mt = 0; mt < 2; ++mt) {
        #pragma unroll
        for (int nt = 0; nt < 2; ++nt) {
            const int col = bn + wn + nt * 16 + l16;
            const float bv = bias ? bias[col] : 0.0f;
            #pragma unroll
            for (int v = 0; v < 8; ++v) {
                const int row = bm + wm + mt * 16 + v + 8 * hi;
                Y[(size_t)row * Nout + col] = acc[mt][nt][v] + bv;
            }
        }
    }
}

// ---------------------------------------------------------------------------
// Flash attention, one wave (32 threads) per (head, 16-query block).
// q,k,v: [4096, 1024] fp32 (head h occupies cols h*64..h*64+63).
// Output scattered into the reference's permuted layout:
//   att[h,n,d] -> att_perm[h*256 + 4*d + (n>>10), n & 1023]
// ---------------------------------------------------------------------------
__global__ __launch_bounds__(32) void attn_flash_f32(
    const float* __restrict__ q, const float* __restrict__ k,
    const float* __restrict__ v, float* __restrict__ att_perm) {
    const int head = blockIdx.y;         // 0..15
    const int qb   = blockIdx.x * 16;    // query block base (16 rows)
    const int lane = threadIdx.x;
    const int l16  = lane & 15;
    const int hi   = lane >> 4;

    __shared__ float S[16][17];
    __shared__ float P[16][17];
    __shared__ float mrow[16], lrow[16], alpha[16];

    const float scale = 0.125f;  // 1/sqrt(64)
    const int hcol = head * 64;

    // Q fragments (A layout), pre-scaled. 16 k-steps covering DH=64.
    v2f qf[16];
    {
        const int qrow = qb + l16;
        #pragma unroll
        for (int s = 0; s < 16; ++s) {
            const int d = 4 * s + 2 * hi;
            v2f t = *(const v2f*)&q[(size_t)qrow * 1024 + hcol + d];
            qf[s].x = t.x * scale;
            qf[s].y = t.y * scale;
        }
    }

    if (lane < 16) { mrow[lane] = -1e30f; lrow[lane] = 0.0f; }
    __syncthreads();

    const v8f vzero = {};
    v8f o[4];
    o[0] = vzero; o[1] = vzero; o[2] = vzero; o[3] = vzero;

    for (int j = 0; j < 4096; j += 16) {
        // ---- S tile = (Q*scale) @ K^T : B frag lane -> key = j+l16 ----
        v8f s8 = vzero;
        #pragma unroll
        for (int s = 0; s < 16; ++s) {
            const int d = 4 * s + 2 * hi;
            v2f kf = *(const v2f*)&k[(size_t)(j + l16) * 1024 + hcol + d];
            s8 = wmma_f32(qf[s], kf, s8);
        }
        // D-layout -> LDS (row = v + 8*hi, col = l16)
        #pragma unroll
        for (int vv = 0; vv < 8; ++vv) S[vv + 8 * hi][l16] = s8[vv];
        __syncthreads();

        // ---- online softmax per query row (lanes 0..15) ----
        if (lane < 16) {
            const float m_old = mrow[lane];
            float mx = m_old;
            #pragma unroll
            for (int c = 0; c < 16; ++c) mx = fmaxf(mx, S[lane][c]);
            const float al = __expf(m_old - mx);
            float sum = 0.0f;
            #pragma unroll
            for (int c = 0; c < 16; ++c) {
                const float p = __expf(S[lane][c] - mx);
                P[lane][c] = p;
                sum += p;
            }
            mrow[lane]  = mx;
            lrow[lane]  = lrow[lane] * al + sum;
            alpha[lane] = al;
        }
        __syncthreads();

        // ---- rescale O accumulators by alpha[row] ----
        #pragma unroll
        for (int t = 0; t < 4; ++t)
            #pragma unroll
            for (int vv = 0; vv < 8; ++vv) o[t][vv] *= alpha[vv + 8 * hi];

        // ---- O += P @ V : P in A layout from LDS, V in B layout ----
        v2f pf[4];
        #pragma unroll
        for (int s = 0; s < 4; ++s)
            pf[s] = *(const v2f*)&P[l16][4 * s + 2 * hi];
        #pragma unroll
        for (int t = 0; t < 4; ++t) {
            #pragma unroll
            for (int s = 0; s < 4; ++s) {
                const int kk = 4 * s + 2 * hi;
                v2f vf;
                vf.x = v[(size_t)(j + kk) * 1024 + hcol + t * 16 + l16];
                vf.y = v[(size_t)(j + kk + 1) * 1024 + hcol + t * 16 + l16];
                o[t] = wmma_f32(pf[s], vf, o[t]);
            }
        }
        __syncthreads();
    }

    // ---- finalize (divide by l) and scatter into permuted layout ----
    #pragma unroll
    for (int t = 0; t < 4; ++t) {
        #pragma unroll
        for (int vv = 0; vv < 8; ++vv) {
            const int m = vv + 8 * hi;     // query row within tile
            const int n = qb + m;          // token index
            const int d = t * 16 + l16;    // head-dim index
            const float val = o[t][vv] / lrow[m];
            const int row = head * 256 + 4 * d + (n >> 10);
            const int col = n & 1023;
            att_perm[(size_t)row * 1024 + col] = val;
        }
    }
}

// ---------------------------------------------------------------------------
// out[row] = layernorm(a[row] + b[row]) * g + beta ; row length 1024,
// one 256-thread block per row (4 elements/thread).
// ---------------------------------------------------------------------------
__global__ __launch_bounds__(256) void add_layernorm_f32(
    const float* __restrict__ a, const float* __restrict__ b,
    const float* __restrict__ g, const float* __restrict__ beta,
    float* __restrict__ out) {
    const int row = blockIdx.x;
    const int tid = threadIdx.x;
    __shared__ float red[256];

    float vals[4];
    float sum = 0.0f;
    #pragma unroll
    for (int i = 0; i < 4; ++i) {
        const int c = tid + i * 256;
        vals[i] = a[(size_t)row * 1024 + c] + b[(size_t)row * 1024 + c];
        sum += vals[i];
    }
    red[tid] = sum;
    __syncthreads();
    #pragma unroll
    for (int s = 128; s > 0; s >>= 1) {
        if (tid < s) red[tid] += red[tid + s];
        __syncthreads();
    }
    const float mu = red[0] * (1.0f / 1024.0f);
    __syncthreads();

    float vs = 0.0f;
    #pragma unroll
    for (int i = 0; i < 4; ++i) {
        const float d = vals[i] - mu;
        vs += d * d;
    }
    red[tid] = vs;
    __syncthreads();
    #pragma unroll
    for (int s = 128; s > 0; s >>= 1) {
        if (tid < s) red[tid] += red[tid + s];
        __syncthreads();
    }
    const float rstd = rsqrtf(red[0] * (1.0f / 1024.0f) + 1e-5f);

    #pragma unroll
    for (int i = 0; i < 4; ++i) {
        const int c = tid + i * 256;
        out[(size_t)row * 1024 + c] = (vals[i] - mu) * rstd * g[c] + beta[c];
    }
}

// ---------------------------------------------------------------------------
extern "C" void kernel_launch(void* const* d_in, const int* in_sizes, int n_in,
                              void* d_out, int out_size, void* d_ws,
                              size_t ws_size, hipStream_t stream) {
    const float* x     = (const float*)d_in[0];
    const float* wq    = (const float*)d_in[1];
    const float* wk    = (const float*)d_in[2];
    const float* wv    = (const float*)d_in[3];
    const float* ln1_g = (const float*)d_in[4];
    const float* ln1_b = (const float*)d_in[5];
    const float* w1    = (const float*)d_in[6];
    const float* b1    = (const float*)d_in[7];
    const float* ln2_g = (const float*)d_in[8];
    const float* ln2_b = (const float*)d_in[9];

    const size_t NM = (size_t)4096 * 1024;
    float* q   = (float*)d_ws;      // 16 MB
    float* k   = q + NM;            // 16 MB
    float* v   = k + NM;            // 16 MB
    float* att = v + NM;            // 16 MB
    float* h1  = q;                 // reuse after attention
    float* ffn = k;                 // reuse after LN1

    const dim3 gemm_grid(4096 / 128, 1024 / 64);  // 32 x 16

    gemm_xwt_f32<<<gemm_grid, 256, 0, stream>>>(x, wq, nullptr, q, 4096, 1024, 1024);
    gemm_xwt_f32<<<gemm_grid, 256, 0, stream>>>(x, wk, nullptr, k, 4096, 1024, 1024);
    gemm_xwt_f32<<<gemm_grid, 256, 0, stream>>>(x, wv, nullptr, v, 4096, 1024, 1024);

    attn_flash_f32<<<dim3(256, 16), 32, 0, stream>>>(q, k, v, att);

    add_layernorm_f32<<<4096, 256, 0, stream>>>(att, x, ln1_g, ln1_b, h1);

    gemm_xwt_f32<<<gemm_grid, 256, 0, stream>>>(h1, w1, b1, ffn, 4096, 1024, 1024);

    add_layernorm_f32<<<4096, 256, 0, stream>>>(ffn, h1, ln2_g, ln2_b, (float*)d_out);
}